// MultiModalGroupFusion_24386824307274
// MI455X (gfx1250) — compile-verified
//
#include <hip/hip_runtime.h>
#include <hip/hip_bf16.h>

typedef __attribute__((ext_vector_type(2))) float v2f;
typedef __attribute__((ext_vector_type(8))) float v8f;

#define K1C 8192
#define K2C 8192
#define DOM1 (1 << 17)
#define POOL_RADF 0.1f
#define INT_MINV ((int)(-2147483647 - 1))
#define MT 2  // point tiles per wave in the WMMA argmin

// ---- helpers: order-preserving float<->uint key (for atomic min/max) ----
__device__ __forceinline__ unsigned fkey(float f) {
  unsigned u = __float_as_uint(f);
  return (u & 0x80000000u) ? ~u : (u | 0x80000000u);
}
__device__ __forceinline__ float funkey(unsigned k) {
  return (k & 0x80000000u) ? __uint_as_float(k ^ 0x80000000u) : __uint_as_float(~k);
}

// ---- init: minmax sentinels + INT_MIN segment-max identities ----
__global__ void init_kernel(unsigned* mm, int* spb, int* bmax, int k1, int k2) {
  int i = blockIdx.x * blockDim.x + threadIdx.x;
  if (i < 6) mm[i] = (i < 3) ? 0xFFFFFFFFu : 0u;
  if (i < k1) spb[i] = INT_MINV;
  if (i < k2) bmax[i] = INT_MINV;
}

// ---- global min/max of all positions ----
__global__ void minmax_kernel(const float* __restrict__ p1, const float* __restrict__ p2,
                              int N1, int NT, unsigned* mm) {
  int i = blockIdx.x * blockDim.x + threadIdx.x;
  if (i >= NT) return;
  const float* p = (i < N1) ? (p1 + 3 * (size_t)i) : (p2 + 3 * (size_t)(i - N1));
#pragma unroll
  for (int d = 0; d < 3; ++d) {
    unsigned k = fkey(p[d]);
    atomicMin(&mm[d], k);
    atomicMax(&mm[3 + d], k);
  }
}

// ---- voxel grid parameters ----
__global__ void params_kernel(const unsigned* mm, float* startv, int* ncv) {
  if (blockIdx.x == 0 && threadIdx.x == 0) {
    for (int d = 0; d < 3; ++d) {
      float mn = funkey(mm[d]);
      float mx = funkey(mm[3 + d]);
      float st = mn - POOL_RADF * 0.5f;
      float en = mx + POOL_RADF * 0.5f;
      startv[d] = st;
      ncv[d] = (int)floorf((en - st) / POOL_RADF) + 1;
    }
  }
}

// ---- cell id per point + presence mark ----
__global__ void cell_kernel(const float* __restrict__ p1, const float* __restrict__ p2,
                            const int* __restrict__ b1, const int* __restrict__ b2,
                            const float* __restrict__ startv, const int* __restrict__ ncv,
                            int N1, int NT, int* __restrict__ cellArr, int* __restrict__ present) {
  int i = blockIdx.x * blockDim.x + threadIdx.x;
  if (i >= NT) return;
  float x, y, z; int b;
  if (i < N1) { const float* p = p1 + 3 * (size_t)i; x = p[0]; y = p[1]; z = p[2]; b = b1[i]; }
  else { int j = i - N1; const float* p = p2 + 3 * (size_t)j; x = p[0]; y = p[1]; z = p[2]; b = b2[j]; }
  int nc0 = ncv[0], nc1 = ncv[1], nc2 = ncv[2];
  int vx = (int)floorf((x - startv[0]) / POOL_RADF);
  int vy = (int)floorf((y - startv[1]) / POOL_RADF);
  int vz = (int)floorf((z - startv[2]) / POOL_RADF);
  vx = vx < 0 ? 0 : (vx > nc0 - 1 ? nc0 - 1 : vx);
  vy = vy < 0 ? 0 : (vy > nc1 - 1 ? nc1 - 1 : vy);
  vz = vz < 0 ? 0 : (vz > nc2 - 1 ? nc2 - 1 : vz);
  long long cell = ((((long long)b * nc0 + vx) * nc1 + vy) * nc2 + vz);
  int c = (cell < 0) ? 0 : (cell >= DOM1 ? DOM1 - 1 : (int)cell);
  cellArr[i] = c;
  present[c] = 1;
}

// ---- single-workgroup exclusive scan over n (n % 1024 == 0) ----
__global__ void exscan1024(const int* __restrict__ in, int* __restrict__ out, int n) {
  __shared__ int sh[1024];
  int t = threadIdx.x;
  int len = n >> 10;
  int base = t * len;
  int s = 0;
  for (int j = 0; j < len; ++j) s += in[base + j];
  sh[t] = s;
  __syncthreads();
  for (int off = 1; off < 1024; off <<= 1) {
    int v = (t >= off) ? sh[t - off] : 0;
    __syncthreads();
    sh[t] += v;
    __syncthreads();
  }
  int run = sh[t] - s;
  for (int j = 0; j < len; ++j) {
    int v = in[base + j];
    out[base + j] = run;
    run += v;
  }
}

// ---- segment accumulate pass 1 (voxel centroids) ----
__global__ void seg1_kernel(const float* __restrict__ p1, const float* __restrict__ p2,
                            const int* __restrict__ b1, const int* __restrict__ b2,
                            const int* __restrict__ cellArr, const int* __restrict__ rank1,
                            float* __restrict__ sumpos, float* __restrict__ cnt1,
                            int* __restrict__ spb, int N1, int NT) {
  int i = blockIdx.x * blockDim.x + threadIdx.x;
  if (i >= NT) return;
  float x, y, z; int b;
  if (i < N1) { const float* p = p1 + 3 * (size_t)i; x = p[0]; y = p[1]; z = p[2]; b = b1[i]; }
  else { int j = i - N1; const float* p = p2 + 3 * (size_t)j; x = p[0]; y = p[1]; z = p[2]; b = b2[j]; }
  int k = rank1[cellArr[i]];
  atomicAdd(&sumpos[3 * k + 0], x);
  atomicAdd(&sumpos[3 * k + 1], y);
  atomicAdd(&sumpos[3 * k + 2], z);
  atomicAdd(&cnt1[k], 1.0f);
  atomicMax(&spb[k], b);
}

// ---- finalize centroids: sp4 = (cx, cy, cz, -0.5*|c|^2)  [cc folded into WMMA K3] ----
__global__ void fin1_kernel(const float* __restrict__ sumpos, const float* __restrict__ cnt1,
                            const int* __restrict__ spb, float* __restrict__ sp4,
                            int* __restrict__ cbv, int K) {
  int k = blockIdx.x * blockDim.x + threadIdx.x;
  if (k >= K) return;
  float m = fmaxf(cnt1[k], 1.0f);
  float sx = sumpos[3 * k + 0] / m;
  float sy = sumpos[3 * k + 1] / m;
  float sz = sumpos[3 * k + 2] / m;
  sp4[4 * k + 0] = sx;
  sp4[4 * k + 1] = sy;
  sp4[4 * k + 2] = sz;
  sp4[4 * k + 3] = -0.5f * (sx * sx + sy * sy + sz * sz);
  cbv[k] = spb[k];
}

// ---- WMMA masked nearest-centroid argmin ----
// One wave owns MT 16-point tiles. A = (x,y | z,1), B = (cx,cy | cz,-cc/2)
// => acc = p.c - cc/2, d2 = fma(-2, acc, pp). Branchless cndmask min-updates.
__global__ void __launch_bounds__(128) argmin_kernel(
    const float* __restrict__ p1, const float* __restrict__ p2,
    const int* __restrict__ b1, const int* __restrict__ b2,
    const float* __restrict__ sp4, const int* __restrict__ cbv,
    int* __restrict__ c2a, int N1, int NT, int K) {
  int lane = threadIdx.x & 31;
  int wave = blockIdx.x * (blockDim.x >> 5) + (threadIdx.x >> 5);
  int base = wave * (16 * MT);
  if (base >= NT) return;
  int col = lane & 15;
  int hi = lane >> 4;

  v2f A[MT];
  float ppv[MT][8];
  int btv[MT][8];
#pragma unroll
  for (int t = 0; t < MT; ++t) {
    int m = base + t * 16 + col;
    int mc = m < NT ? m : NT - 1;
    float px, py, pz; int pb;
    if (mc < N1) { const float* p = p1 + 3 * (size_t)mc; px = p[0]; py = p[1]; pz = p[2]; pb = b1[mc]; }
    else { int j = mc - N1; const float* p = p2 + 3 * (size_t)j; px = p[0]; py = p[1]; pz = p[2]; pb = b2[j]; }
    float pp = px * px + py * py + pz * pz;
    A[t][0] = (hi == 0) ? px : pz;
    A[t][1] = (hi == 0) ? py : 1.0f;  // K3 element multiplies B's -cc/2 row
#pragma unroll
    for (int v = 0; v < 8; ++v) {
      int src = v + 8 * hi;
      ppv[t][v] = __shfl(pp, src, 32);
      btv[t][v] = __shfl(pb, src, 32);
    }
  }

  float best[MT][8];
  int bidx[MT][8];
#pragma unroll
  for (int t = 0; t < MT; ++t)
#pragma unroll
    for (int v = 0; v < 8; ++v) { best[t][v] = __builtin_inff(); bidx[t][v] = 0; }

  for (int n0 = 0; n0 < K; n0 += 16) {
    int n = n0 + col;
    v2f B = *(const v2f*)(sp4 + 4 * (size_t)n + 2 * hi);  // unconditional b64 load
    int cbn = cbv[n];
#pragma unroll
    for (int t = 0; t < MT; ++t) {
      v8f acc = {};
      acc = __builtin_amdgcn_wmma_f32_16x16x4_f32(false, A[t], false, B, (short)0, acc,
                                                  false, false);
#pragma unroll
      for (int v = 0; v < 8; ++v) {
        float d2 = fmaf(-2.0f, acc[v], ppv[t][v]);
        float cand = (cbn == btv[t][v]) ? d2 : __builtin_inff();
        bool upd = cand < best[t][v];
        bidx[t][v] = upd ? n : bidx[t][v];
        best[t][v] = upd ? cand : best[t][v];
      }
    }
  }

  // reduce (min, lowest-index tiebreak) across the 16 lanes of each half
#pragma unroll
  for (int t = 0; t < MT; ++t) {
#pragma unroll
    for (int v = 0; v < 8; ++v) {
      for (int mask = 1; mask < 16; mask <<= 1) {
        float od = __shfl_xor(best[t][v], mask, 32);
        int oi = __shfl_xor(bidx[t][v], mask, 32);
        bool take = (od < best[t][v]) || ((od == best[t][v]) && (oi < bidx[t][v]));
        best[t][v] = take ? od : best[t][v];
        bidx[t][v] = take ? oi : bidx[t][v];
      }
      if (col == 0) {
        int idx = base + t * 16 + v + 8 * hi;
        if (idx < NT) c2a[idx] = bidx[t][v];
      }
    }
  }
}

// ---- unique(c2): presence mark ----
__global__ void mark2_kernel(const int* __restrict__ c2a, int* __restrict__ present2, int NT) {
  int i = blockIdx.x * blockDim.x + threadIdx.x;
  if (i >= NT) return;
  present2[c2a[i]] = 1;
}

// ---- segment accumulate pass 2 (output clusters) ----
__global__ void seg2_kernel(const float* __restrict__ p1, const float* __restrict__ p2,
                            const int* __restrict__ b1, const int* __restrict__ b2,
                            const int* __restrict__ c2a, const int* __restrict__ rank2,
                            float* __restrict__ accpos, float* __restrict__ cntp,
                            int* __restrict__ bmax, float* __restrict__ cntm,
                            int N1, int NT) {
  int i = blockIdx.x * blockDim.x + threadIdx.x;
  if (i >= NT) return;
  float x, y, z; int b;
  if (i < N1) { const float* p = p1 + 3 * (size_t)i; x = p[0]; y = p[1]; z = p[2]; b = b1[i]; }
  else { int j = i - N1; const float* p = p2 + 3 * (size_t)j; x = p[0]; y = p[1]; z = p[2]; b = b2[j]; }
  int k = rank2[c2a[i]];
  atomicAdd(&accpos[3 * k + 0], x);
  atomicAdd(&accpos[3 * k + 1], y);
  atomicAdd(&accpos[3 * k + 2], z);
  atomicAdd(&cntp[k], 1.0f);
  atomicMax(&bmax[k], b);
  int mod = (i < N1) ? 0 : 1;
  atomicAdd(&cntm[2 * k + mod], 1.0f);
}

// ---- feature scatter-accumulate ----
__global__ void feat_kernel(const float* __restrict__ x1, const float* __restrict__ x2,
                            const int* __restrict__ c2a, const int* __restrict__ rank2,
                            float* __restrict__ accx, int N1, int F1, int N2, int F2) {
  long long tid = (long long)blockIdx.x * blockDim.x + threadIdx.x;
  long long n1f = (long long)N1 * F1;
  long long tot = n1f + (long long)N2 * F2;
  if (tid >= tot) return;
  int FT = F1 + F2;
  if (tid < n1f) {
    int i = (int)(tid / F1);
    int f = (int)(tid % F1);
    int k = rank2[c2a[i]];
    atomicAdd(&accx[(size_t)k * FT + f], x1[tid]);
  } else {
    long long t2 = tid - n1f;
    int i = (int)(t2 / F2);
    int f = (int)(t2 % F2);
    int k = rank2[c2a[N1 + i]];
    atomicAdd(&accx[(size_t)k * FT + F1 + f], x2[t2]);
  }
}

// ---- finalize outputs: x_out | pos_out | batch_out (flat, float32) ----
__global__ void final_kernel(const float* __restrict__ accx, const float* __restrict__ cntm,
                             const float* __restrict__ accpos, const float* __restrict__ cntp,
                             const int* __restrict__ bmax, float* __restrict__ out,
                             int K2v, int F1, int FT) {
  int tid = blockIdx.x * blockDim.x + threadIdx.x;
  if (tid >= K2v * FT) return;
  int k = tid / FT;
  int f = tid % FT;
  int mod = (f < F1) ? 0 : 1;
  float c = cntm[2 * k + mod];
  out[(size_t)k * FT + f] = (1.0f + accx[(size_t)k * FT + f]) / fmaxf(c, 1.0f);
  if (f == 0) {
    float* out_pos = out + (size_t)K2v * FT;
    float* out_b = out_pos + (size_t)K2v * 3;
    float pc = cntp[k];
    float m = fmaxf(pc, 1.0f);
    out_pos[3 * k + 0] = accpos[3 * k + 0] / m;
    out_pos[3 * k + 1] = accpos[3 * k + 1] / m;
    out_pos[3 * k + 2] = accpos[3 * k + 2] / m;
    out_b[k] = (pc > 0.0f) ? (float)bmax[k] : -1.0f;
  }
}

extern "C" void kernel_launch(void* const* d_in, const int* in_sizes, int n_in,
                              void* d_out, int out_size, void* d_ws, size_t ws_size,
                              hipStream_t stream) {
  const float* pos1 = (const float*)d_in[0];
  const float* x1 = (const float*)d_in[1];
  const int* batch1 = (const int*)d_in[2];
  const float* pos2 = (const float*)d_in[3];
  const float* x2 = (const float*)d_in[4];
  const int* batch2 = (const int*)d_in[5];
  int N1 = in_sizes[2];
  int N2 = in_sizes[5];
  int F1 = in_sizes[1] / N1;
  int F2 = in_sizes[4] / N2;
  int NT = N1 + N2;
  int FT = F1 + F2;

  char* ws = (char*)d_ws;
  size_t o = 0;
  auto take = [&](size_t bytes) {
    size_t r = o;
    o += (bytes + 255) & ~(size_t)255;
    return r;
  };

  unsigned* mm   = (unsigned*)(ws + take(6 * 4));
  float* startv  = (float*)(ws + take(3 * 4));
  int* ncv       = (int*)(ws + take(3 * 4));
  int* cellArr   = (int*)(ws + take((size_t)NT * 4));
  int* rank1     = (int*)(ws + take((size_t)DOM1 * 4));
  int* rank2     = (int*)(ws + take((size_t)K1C * 4));
  float* sp4     = (float*)(ws + take((size_t)K1C * 16));
  int* cbv       = (int*)(ws + take((size_t)K1C * 4));
  int* c2a       = (int*)(ws + take((size_t)NT * 4));
  int* spb       = (int*)(ws + take((size_t)K1C * 4));
  int* bmax      = (int*)(ws + take((size_t)K2C * 4));
  // zero-initialized zone (contiguous -> one memset per launch)
  size_t z0 = o;
  int* present1  = (int*)(ws + take((size_t)DOM1 * 4));
  int* present2  = (int*)(ws + take((size_t)K1C * 4));
  float* sumpos  = (float*)(ws + take((size_t)K1C * 12));
  float* cnt1    = (float*)(ws + take((size_t)K1C * 4));
  float* accpos  = (float*)(ws + take((size_t)K2C * 12));
  float* cntp    = (float*)(ws + take((size_t)K2C * 4));
  float* cntm    = (float*)(ws + take((size_t)K2C * 8));
  float* accx    = (float*)(ws + take((size_t)K2C * (size_t)FT * 4));
  size_t zlen = o - z0;

  hipMemsetAsync(ws + z0, 0, zlen, stream);

  const int tpb = 256;
  int initN = (K1C > K2C ? K1C : K2C);
  init_kernel<<<(initN + tpb - 1) / tpb, tpb, 0, stream>>>(mm, spb, bmax, K1C, K2C);
  minmax_kernel<<<(NT + tpb - 1) / tpb, tpb, 0, stream>>>(pos1, pos2, N1, NT, mm);
  params_kernel<<<1, 32, 0, stream>>>(mm, startv, ncv);
  cell_kernel<<<(NT + tpb - 1) / tpb, tpb, 0, stream>>>(pos1, pos2, batch1, batch2, startv, ncv,
                                                        N1, NT, cellArr, present1);
  exscan1024<<<1, 1024, 0, stream>>>(present1, rank1, DOM1);
  seg1_kernel<<<(NT + tpb - 1) / tpb, tpb, 0, stream>>>(pos1, pos2, batch1, batch2, cellArr, rank1,
                                                        sumpos, cnt1, spb, N1, NT);
  fin1_kernel<<<(K1C + tpb - 1) / tpb, tpb, 0, stream>>>(sumpos, cnt1, spb, sp4, cbv, K1C);
  int waves = (NT + 16 * MT - 1) / (16 * MT);
  int blocks = (waves + 3) / 4;
  argmin_kernel<<<blocks, 128, 0, stream>>>(pos1, pos2, batch1, batch2, sp4, cbv, c2a,
                                            N1, NT, K1C);
  mark2_kernel<<<(NT + tpb - 1) / tpb, tpb, 0, stream>>>(c2a, present2, NT);
  exscan1024<<<1, 1024, 0, stream>>>(present2, rank2, K1C);
  seg2_kernel<<<(NT + tpb - 1) / tpb, tpb, 0, stream>>>(pos1, pos2, batch1, batch2, c2a, rank2,
                                                        accpos, cntp, bmax, cntm, N1, NT);
  long long featTot = (long long)N1 * F1 + (long long)N2 * F2;
  feat_kernel<<<(int)((featTot + tpb - 1) / tpb), tpb, 0, stream>>>(x1, x2, c2a, rank2, accx,
                                                                    N1, F1, N2, F2);
  final_kernel<<<(K2C * FT + tpb - 1) / tpb, tpb, 0, stream>>>(accx, cntm, accpos, cntp, bmax,
                                                               (float*)d_out, K2C, F1, FT);
}